// BundleAdjustmentModel_81157702025654
// MI455X (gfx1250) — compile-verified
//
#include <hip/hip_runtime.h>
#include <hip/hip_bf16.h>

typedef __attribute__((ext_vector_type(2))) float v2f;
typedef __attribute__((ext_vector_type(8))) float v8f;

#define NPTS   200000
#define NVIEWS 128

constexpr float CXc   = 512.0f;
constexpr float CYc   = 512.0f;
constexpr float ZMAXc = -0.0001f;

constexpr int VG      = 5;                        // views per WMMA (3 cols each -> 15/16 cols)
constexpr int NGROUPS = (NVIEWS + VG - 1) / VG;   // 26
constexpr int PTILE   = 16;                       // points per WMMA tile
constexpr int TILES   = NPTS / PTILE;             // 12500 (exact)
constexpr int WPB     = 8;                        // waves per block (256 threads, wave32)

// Fast reciprocal: v_rcp_f32 + one Newton-Raphson step (~1 ulp, 3 VALU ops
// instead of the ~10-op IEEE div_scale/div_fmas/div_fixup sequence).
__device__ __forceinline__ float fast_rcp(float x) {
#if __has_builtin(__builtin_amdgcn_rcpf)
  float r = __builtin_amdgcn_rcpf(x);
  return r * __builtin_fmaf(-x, r, 2.0f);
#else
  return 1.0f / x;
#endif
}

// ---------------------------------------------------------------------------
// Kernel 1: per-view record: rec[v*12 + i*4 + k] = R[i][k] (k<3), rec[..+3]=t[i]
// R = Rx(x) @ Ry(y) @ Rz(z)  (matches euler_angles_to_matrix in the reference)
// ---------------------------------------------------------------------------
__global__ void view_setup_kernel(const float* __restrict__ euler,
                                  const float* __restrict__ trans,
                                  float* __restrict__ rec) {
  int v = threadIdx.x + blockIdx.x * blockDim.x;
  if (v >= NVIEWS) return;
  float ex = euler[v * 3 + 0], ey = euler[v * 3 + 1], ez = euler[v * 3 + 2];
  float c1 = cosf(ex), s1 = sinf(ex);
  float c2 = cosf(ey), s2 = sinf(ey);
  float c3 = cosf(ez), s3 = sinf(ez);

  float* r = rec + v * 12;
  // Row 0
  r[0]  = c2 * c3;
  r[1]  = -c2 * s3;
  r[2]  = s2;
  r[3]  = trans[v * 3 + 0];
  // Row 1
  r[4]  = c1 * s3 + s1 * s2 * c3;
  r[5]  = c1 * c3 - s1 * s2 * s3;
  r[6]  = -s1 * c2;
  r[7]  = trans[v * 3 + 1];
  // Row 2
  r[8]  = s1 * s3 - c1 * s2 * c3;
  r[9]  = s1 * c3 + c1 * s2 * s3;
  r[10] = c1 * c2;
  r[11] = trans[v * 3 + 2];
}

// ---------------------------------------------------------------------------
// Kernel 2: one wave = one WMMA = 16 points x 5 views (homogeneous transform),
// then per-lane projection from LDS-staged camera coords, NT float2 stores.
// ---------------------------------------------------------------------------
__global__ void __launch_bounds__(WPB * 32)
project_kernel(const float* __restrict__ pts,     // [NPTS][3]
               const float* __restrict__ rec,     // [NVIEWS][12]
               const float* __restrict__ focal,   // [1]
               float* __restrict__ out) {         // [NVIEWS][NPTS][2]
  __shared__ float lds[WPB][PTILE][17];           // pad 17: stride-16 re-read hits distinct banks

  const int lane = threadIdx.x & 31;
  const int w    = threadIdx.x >> 5;
  const int wave = blockIdx.x * WPB + w;          // 0 .. 325000-1 (exact grid)
  const int tile = wave / NGROUPS;                // point tile, shared by 26 waves in a row
  const int vg   = wave % NGROUPS;                // view group

  const int base = tile * PTILE;
  const int hi   = lane >> 4;                     // 0: lanes 0-15, 1: lanes 16-31
  const int col  = lane & 15;                     // A: point row M; B/D: column N

  // ---- A matrix (16x4 f32): rows = [x y z 1] per point ---------------------
  // ISA layout: VGPR0 = K=0 (lo) / K=2 (hi), VGPR1 = K=1 (lo) / K=3 (hi)
  const float* p = pts + 3 * (base + col);
  v2f a;
  a.x = hi ? p[2] : p[0];
  a.y = hi ? 1.0f : p[1];

  // ---- B matrix (4x16 f32): column c = (view vg*5 + c/3, component c%3) ----
  // B[k][c] = rec[view*12 + comp*4 + k]; VGPR0 = K=0/2, VGPR1 = K=1/3
  const int jv   = col / 3;
  const int comp = col - 3 * jv;
  const int view = vg * VG + jv;
  v2f b = {0.0f, 0.0f};
  if (col < 15 && view < NVIEWS) {
    const float* rp = rec + view * 12 + comp * 4 + hi * 2;
    b.x = rp[0];
    b.y = rp[1];
  }

  // ---- D = A x B (+0): camera coords for 16 points x 5 views ---------------
  v8f c = {};
  c = __builtin_amdgcn_wmma_f32_16x16x4_f32(
      /*neg_a=*/false, a, /*neg_b=*/false, b,
      /*c_mod=*/(short)0, c, /*reuse_a=*/false, /*reuse_b=*/false);

  // ---- stage D through LDS: lds[w][M][N], M = r + hi*8, N = col ------------
#pragma unroll
  for (int r = 0; r < 8; ++r)
    lds[w][(hi << 3) + r][col] = c[r];
  __syncthreads();

  // ---- projection: 5 views x 16 points = 80 (u,v) pairs per wave -----------
  const float f = focal[0];
#pragma unroll
  for (int it = 0; it < 3; ++it) {
    int idx = lane + 32 * it;                     // 0..95, use 0..79
    if (idx < VG * PTILE) {
      int j = idx >> 4;                           // view within group
      int m = idx & 15;                           // point within tile
      int vv = vg * VG + j;
      if (vv < NVIEWS) {
        float Xc = lds[w][m][3 * j + 0];
        float Yc = lds[w][m][3 * j + 1];
        float Zc = fminf(lds[w][m][3 * j + 2], ZMAXc);
        float inv = fast_rcp(Zc);                 // one rcp+NR, shared by u and v
        v2f uv;
        uv.x = __builtin_fmaf(-f * Xc, inv, CXc);
        uv.y = __builtin_fmaf( f * Yc, inv, CYc);
        // contiguous per view: 16 lanes cover a 128B line; NT: 205MB > L2
        __builtin_nontemporal_store(
            uv, (v2f*)out + ((size_t)vv * NPTS + base + m));
      }
    }
  }
}

// ---------------------------------------------------------------------------
extern "C" void kernel_launch(void* const* d_in, const int* in_sizes, int n_in,
                              void* d_out, int out_size, void* d_ws, size_t ws_size,
                              hipStream_t stream) {
  const float* points = (const float*)d_in[0];   // 200000*3
  const float* euler  = (const float*)d_in[1];   // 128*3
  const float* trans  = (const float*)d_in[2];   // 128*3
  const float* focal  = (const float*)d_in[3];   // 1
  float* out = (float*)d_out;                    // 128*200000*2
  float* rec = (float*)d_ws;                     // 128*12 floats = 6144 B scratch

  view_setup_kernel<<<1, NVIEWS, 0, stream>>>(euler, trans, rec);

  const int total_waves = TILES * NGROUPS;       // 325000
  const int blocks = total_waves / WPB;          // 40625 (exact, no tail)
  project_kernel<<<blocks, WPB * 32, 0, stream>>>(points, rec, focal, out);
}